// GCN_56384330662074
// MI455X (gfx1250) — compile-verified
//
#include <hip/hip_runtime.h>
#include <math.h>

typedef __attribute__((ext_vector_type(2))) float        v2f;
typedef __attribute__((ext_vector_type(8))) float        v8f;
typedef __attribute__((ext_vector_type(4))) unsigned int v4u;
typedef __attribute__((ext_vector_type(8))) int          v8i;
typedef __attribute__((ext_vector_type(4))) int          v4i;

#define N_NODES 100000
#define F_IN    128
#define F_HID   64
#define F_OUT   40
#define M_TILES (N_NODES / 16)   // 6250, exact

#if defined(__gfx1250__) && __has_builtin(__builtin_amdgcn_tensor_load_to_lds) && \
    __has_builtin(__builtin_amdgcn_s_wait_tensorcnt)
#define HAVE_TDM 1
#else
#define HAVE_TDM 0
#endif

// ---------------------------------------------------------------------------
// Utility fills
// ---------------------------------------------------------------------------
__global__ __launch_bounds__(256) void fill_f32(float* __restrict__ p, float val, int n) {
    int i = blockIdx.x * blockDim.x + threadIdx.x;
    if (i < n) p[i] = val;
}

// deg[dst]++ per edge (deg pre-filled with 1.0 for self loops)
__global__ __launch_bounds__(256) void deg_accum(const long long* __restrict__ dst,
                                                 float* __restrict__ deg, int E) {
    int e = blockIdx.x * blockDim.x + threadIdx.x;
    if (e + 4096 < E) __builtin_prefetch(&dst[e + 4096], 0, 0);
    if (e < E) atomicAdd(&deg[(int)dst[e]], 1.0f);
}

// in-place deg -> dinv = rsqrt(deg)
__global__ __launch_bounds__(256) void make_dinv(float* __restrict__ deg, int n) {
    int i = blockIdx.x * blockDim.x + threadIdx.x;
    if (i < n) {
        float d = deg[i];
        deg[i] = (d > 0.0f) ? rsqrtf(d) : 0.0f;
    }
}

// ---------------------------------------------------------------------------
// WMMA fp32 GEMM with LDS-staged, zero-padded B.
//   C[M x NCOLS] = A[M x K] * B[K x NCOLS]   (row-major, M multiple of 16)
// Block = 256 threads = 8 waves; each wave owns one 16-row M-tile and
// computes ALL column tiles (NTILES), reusing one A fragment per k-step.
//
// B staging: preferred path is the Tensor Data Mover. One wave issues
// TENSOR_LOAD_TO_LDS with tile_dim0 = NPAD > tensor_dim0 = NCOLS; TDM OOB
// reads return zero while the LDS cursor advances contiguously, yielding a
// zero-padded K x NPAD image for free. Completion via s_wait_tensorcnt 0,
// visibility via the workgroup barrier. Fallback: cooperative VALU loads.
//
// VGPR layouts per CDNA5 ISA 7.12.2 (32-bit, wave32):
//   A 16x4 : a[j] = A[lane&15][2*(lane>>4)+j]
//   B 4x16 : b[j] = B[2*(lane>>4)+j][lane&15]
//   C 16x16: c[r] = C[r + 8*(lane>>4)][lane&15]
// ---------------------------------------------------------------------------
template <int K, int NCOLS, int NTILES>
__global__ __launch_bounds__(256) void wmma_gemm_f32_lds(
    const float* __restrict__ A, const float* __restrict__ B,
    float* __restrict__ C, int mtiles)
{
    constexpr int NPAD = NTILES * 16;
    __shared__ float Bs[K * NPAD];

    const int tid = threadIdx.x;

#if HAVE_TDM
    if (tid < 32) {                              // one wave issues the DMA
        unsigned long long ga = (unsigned long long)(uintptr_t)B;
        unsigned int       la = (unsigned int)(uintptr_t)Bs;   // LDS byte address

        // D# group 0 (ISA 8.3): count=1 | lds_addr | global_addr[56:0] | type=2
        v4u g0;
        g0.x = 1u;                                                  // count=1, user
        g0.y = la;                                                  // lds_addr
        g0.z = (unsigned int)ga;                                    // global[31:0]
        g0.w = (unsigned int)((ga >> 32) & 0x01FFFFFFull) | (2u << 30); // type=2

        // D# group 1 (ISA 8.4), 2D tensor, data_size=2 (4 bytes):
        v8i g1;
        g1[0] = 0x00020000;                      // wg_mask=0, data_size=2
        g1[1] = (int)((NCOLS & 0xFFFF) << 16);   // tensor_dim0 (bits 79:48 lo)
        g1[2] = (int)((K & 0xFFFF) << 16);       // tensor_dim1 (bits 111:80 lo)
        g1[3] = (int)((NPAD & 0xFFFF) << 16);    // tile_dim0   (bits 127:112)
        g1[4] = (int)(K & 0xFFFF);               // tile_dim1   (bits 143:128)
        g1[5] = (int)NCOLS;                      // tensor_dim0_stride lo 32
        g1[6] = 0;                               // stride hi / dim1_stride lo
        g1[7] = 0;

        v4i gz = {0, 0, 0, 0};                   // groups 2/3 unused (2D)
#if __clang_major__ >= 23
        v8i gz8 = {0, 0, 0, 0, 0, 0, 0, 0};
        __builtin_amdgcn_tensor_load_to_lds(g0, g1, gz, gz, gz8, 0);
#else
        __builtin_amdgcn_tensor_load_to_lds(g0, g1, gz, gz, 0);
#endif
        __builtin_amdgcn_s_wait_tensorcnt(0);
    }
#else
    // Cooperative stage of B into LDS (zero-pad columns >= NCOLS)
    for (int idx = tid; idx < K * NPAD; idx += 256) {
        int k = idx / NPAD;
        int c = idx - k * NPAD;
        Bs[idx] = (c < NCOLS) ? B[k * NCOLS + c] : 0.0f;
    }
#endif
    __syncthreads();

    const int mt = blockIdx.x * 8 + (tid >> 5);
    if (mt >= mtiles) return;                 // wave-uniform guard

    const int lane = tid & 31;
    const int l16  = lane & 15;
    const int lh   = lane >> 4;               // half-wave select

    const float* __restrict__ arow = A + (size_t)(mt * 16 + l16) * K;

    v8f acc[NTILES] = {};

    #pragma unroll 4
    for (int k0 = 0; k0 < K; k0 += 4) {
        const int ka = k0 + 2 * lh;
        v2f a;
        a.x = arow[ka];
        a.y = arow[ka + 1];
        #pragma unroll
        for (int t = 0; t < NTILES; ++t) {
            v2f b;
            b.x = Bs[ka       * NPAD + t * 16 + l16];
            b.y = Bs[(ka + 1) * NPAD + t * 16 + l16];
            // (neg_a, A, neg_b, B, c_mod, C, reuse_a, reuse_b)
            acc[t] = __builtin_amdgcn_wmma_f32_16x16x4_f32(
                false, a, false, b, (short)0, acc[t], false, false);
        }
    }

    const int rbase = mt * 16 + 8 * lh;
    #pragma unroll
    for (int t = 0; t < NTILES; ++t) {
        const int col = t * 16 + l16;
        if (col < NCOLS) {                    // compile-time true except last tile
            #pragma unroll
            for (int r = 0; r < 8; ++r)
                C[(size_t)(rbase + r) * NCOLS + col] = acc[t][r];
        }
    }
}

// ---------------------------------------------------------------------------
// Edge scatter: one WAVE per edge. Edge indices + norm loaded once per wave;
// lanes cover the feature dimension with float2 loads and two global float
// atomics each (targets are L2-resident: agg fits in the 192 MB L2).
// ---------------------------------------------------------------------------
template <int F>
__global__ __launch_bounds__(256) void scatter_edges_wave(
    const long long* __restrict__ src, const long long* __restrict__ dst,
    const float* __restrict__ dinv, const float* __restrict__ xw,
    float* __restrict__ agg, int E)
{
    const int e = blockIdx.x * 8 + (threadIdx.x >> 5);
    if (e >= E) return;                       // wave-uniform guard
    const int lane = threadIdx.x & 31;

    if (e + 1024 < E) {
        __builtin_prefetch(&src[e + 1024], 0, 0);
        __builtin_prefetch(&dst[e + 1024], 0, 0);
    }

    const int s = (int)src[e];
    const int d = (int)dst[e];
    const float norm = dinv[s] * dinv[d];

    const float* __restrict__ xrow = xw  + (size_t)s * F;
    float*       __restrict__ arow = agg + (size_t)d * F;

    for (int f0 = lane * 2; f0 < F; f0 += 64) {
        const float2 v = *(const float2*)(xrow + f0);
        atomicAdd(arow + f0 + 0, v.x * norm);
        atomicAdd(arow + f0 + 1, v.y * norm);
    }
}

// ---------------------------------------------------------------------------
// h = relu(agg1 + xw1*dinv^2 + b1), written in place into agg1
// ---------------------------------------------------------------------------
__global__ __launch_bounds__(256) void hidden_act(
    float* __restrict__ agg1, const float* __restrict__ xw1,
    const float* __restrict__ dinv, const float* __restrict__ b1, int n)
{
    int idx = blockIdx.x * blockDim.x + threadIdx.x;
    if (idx >= n) return;
    int i = idx >> 6;          // / F_HID (64)
    int f = idx & 63;
    float di = dinv[i];
    float h  = agg1[idx] + xw1[idx] * di * di + b1[f];
    agg1[idx] = fmaxf(h, 0.0f);
}

// ---------------------------------------------------------------------------
// out = log_softmax(agg2 + xw2*dinv^2 + b2) per node; 40 values in registers
// ---------------------------------------------------------------------------
__global__ __launch_bounds__(256) void final_logsoftmax(
    const float* __restrict__ agg2, const float* __restrict__ xw2,
    const float* __restrict__ dinv, const float* __restrict__ b2,
    float* __restrict__ out, int n)
{
    int i = blockIdx.x * blockDim.x + threadIdx.x;
    if (i >= n) return;
    float di = dinv[i];
    float d2 = di * di;

    float v[F_OUT];
    float mx = -3.402823466e38f;
    #pragma unroll
    for (int f = 0; f < F_OUT; ++f) {
        float t = agg2[(size_t)i * F_OUT + f] + xw2[(size_t)i * F_OUT + f] * d2 + b2[f];
        v[f] = t;
        mx = fmaxf(mx, t);
    }
    float s = 0.0f;
    #pragma unroll
    for (int f = 0; f < F_OUT; ++f) s += expf(v[f] - mx);
    float lse = mx + logf(s);
    #pragma unroll
    for (int f = 0; f < F_OUT; ++f)
        out[(size_t)i * F_OUT + f] = v[f] - lse;
}

// ---------------------------------------------------------------------------
static inline int cdiv_ll(long long a, long long b) { return (int)((a + b - 1) / b); }

extern "C" void kernel_launch(void* const* d_in, const int* in_sizes, int n_in,
                              void* d_out, int out_size, void* d_ws, size_t ws_size,
                              hipStream_t stream)
{
    (void)n_in; (void)out_size; (void)ws_size;

    const float*     x   = (const float*)d_in[0];
    const long long* ei  = (const long long*)d_in[1];   // int64 edge_index [2, E]
    const float*     W1  = (const float*)d_in[2];
    const float*     b1  = (const float*)d_in[3];
    const float*     W2  = (const float*)d_in[4];
    const float*     b2  = (const float*)d_in[5];

    const int E = in_sizes[1] / 2;
    const long long* src = ei;
    const long long* dst = ei + E;

    // Workspace layout (floats): dinv | xw1 | agg1(->h1) | xw2 | agg2  (~84 MB)
    float* ws   = (float*)d_ws;
    float* dinv = ws;
    float* xw1  = dinv + N_NODES;
    float* agg1 = xw1  + (size_t)N_NODES * F_HID;
    float* xw2  = agg1 + (size_t)N_NODES * F_HID;
    float* agg2 = xw2  + (size_t)N_NODES * F_OUT;

    const int nh = N_NODES * F_HID;   // 6,400,000
    const int no = N_NODES * F_OUT;   // 4,000,000

    // 1) init: deg = 1 (self loop), agg buffers = 0
    fill_f32<<<cdiv_ll(N_NODES, 256), 256, 0, stream>>>(dinv, 1.0f, N_NODES);
    fill_f32<<<cdiv_ll(nh, 256),      256, 0, stream>>>(agg1, 0.0f, nh);
    fill_f32<<<cdiv_ll(no, 256),      256, 0, stream>>>(agg2, 0.0f, no);

    // 2) degrees over targets, then dinv = rsqrt(deg)
    deg_accum<<<cdiv_ll(E, 256), 256, 0, stream>>>(dst, dinv, E);
    make_dinv<<<cdiv_ll(N_NODES, 256), 256, 0, stream>>>(dinv, N_NODES);

    // 3) GEMM1: xw1 = x @ W1  (M=100000, K=128, N=64; 4 col-tiles)
    wmma_gemm_f32_lds<F_IN, F_HID, 4><<<cdiv_ll(M_TILES, 8), 256, 0, stream>>>(
        x, W1, xw1, M_TILES);

    // 4) edge scatter layer 1 (one wave per edge)
    scatter_edges_wave<F_HID><<<cdiv_ll(E, 8), 256, 0, stream>>>(
        src, dst, dinv, xw1, agg1, E);

    // 5) self-loop + bias + ReLU (in place into agg1 -> h1)
    hidden_act<<<cdiv_ll(nh, 256), 256, 0, stream>>>(agg1, xw1, dinv, b1, nh);

    // 6) GEMM2: xw2 = h1 @ W2  (M=100000, K=64, N=40; 3 col-tiles, TDM zero-pad)
    wmma_gemm_f32_lds<F_HID, F_OUT, 3><<<cdiv_ll(M_TILES, 8), 256, 0, stream>>>(
        agg1, W2, xw2, M_TILES);

    // 7) edge scatter layer 2
    scatter_edges_wave<F_OUT><<<cdiv_ll(E, 8), 256, 0, stream>>>(
        src, dst, dinv, xw2, agg2, E);

    // 8) self-loop + bias + log_softmax -> d_out
    final_logsoftmax<<<cdiv_ll(N_NODES, 256), 256, 0, stream>>>(
        agg2, xw2, dinv, b2, (float*)d_out, N_NODES);
}